// RigidTransform_60567628808471
// MI455X (gfx1250) — compile-verified
//
#include <hip/hip_runtime.h>

// ---------------------------------------------------------------------------
// RigidTransform: out = x, with joints 0..7 (of 25) rotated by R(theta),
// theta = weights[0,0].  Pure streaming kernel: ~210 MB HBM traffic, ~25 MFLOP.
// MI455X strategy: wave32 b128 accesses, non-temporal hints (single-use data,
// 210 MB > 192 MB L2), CDNA5 async LDS-DMA (ASYNCcnt) double-buffered prefetch.
// WMMA deliberately NOT used: 2x2 rotation in a 16x16x4 tile is <1% utilized
// and would add layout shuffles to a bandwidth-bound kernel.
// ---------------------------------------------------------------------------

#ifndef __has_builtin
#define __has_builtin(x) 0
#endif

#if __has_builtin(__builtin_amdgcn_global_load_async_to_lds_b128)
#define USE_ASYNC_LDS 1
#else
#define USE_ASYNC_LDS 0
#endif

typedef float v4f __attribute__((ext_vector_type(4)));
typedef int   v4i __attribute__((vector_size(16)));   // builtin's pointee type

#define THREADS 256u
#define TILE    THREADS   // float4 elements per tile (one per thread)

__device__ __forceinline__ void wait_async_le1() {
#if __has_builtin(__builtin_amdgcn_s_wait_asynccnt)
  __builtin_amdgcn_s_wait_asynccnt(1);
#else
  asm volatile("s_wait_asynccnt 0x1" ::: "memory");
#endif
}
__device__ __forceinline__ void wait_async_le0() {
#if __has_builtin(__builtin_amdgcn_s_wait_asynccnt)
  __builtin_amdgcn_s_wait_asynccnt(0);
#else
  asm volatile("s_wait_asynccnt 0x0" ::: "memory");
#endif
}

// Apply the broadcast 2x2 rotation to the (up to) 2 joint-pairs in a float4.
// i = float4 index; pair q0 = 2i has joint j0 = q0 % 25, pair q0+1 has j1.
// Reference math: x' = x*cos + y*sin ; y' = y*cos - x*sin.
__device__ __forceinline__ v4f rotate_pairs(v4f v, unsigned i, float cs, float sn) {
  unsigned q  = 2u * i;
  unsigned j0 = q % 25u;
  unsigned j1 = (j0 + 1u == 25u) ? 0u : (j0 + 1u);
  if (j0 < 8u) {
    float a = v.x, b = v.y;
    v.x = a * cs + b * sn;
    v.y = b * cs - a * sn;
  }
  if (j1 < 8u) {
    float a = v.z, b = v.w;
    v.z = a * cs + b * sn;
    v.w = b * cs - a * sn;
  }
  return v;
}

__global__ __launch_bounds__(THREADS)
void rigid_rot_stream(const v4f* __restrict__ x, const float* __restrict__ w,
                      v4f* __restrict__ out, unsigned n4, unsigned ntiles) {
  const unsigned tid = threadIdx.x;
  const float th = w[0];        // uniform scalar: theta = weights[0,0]
  const float cs = cosf(th);
  const float sn = sinf(th);

#if USE_ASYNC_LDS
  // --- CDNA5 async LDS-DMA double-buffered pipeline -----------------------
  // Each lane DMAs its own 16B slot: no cross-wave sharing -> no barriers,
  // just per-wave s_wait_asynccnt. 8 KB LDS of 320 KB/WGP.
  __shared__ v4f buf[2][TILE];
  const unsigned stride = gridDim.x;
  unsigned t = blockIdx.x;
  int cur = 0;

  if (t < ntiles) {
    unsigned i0 = t * TILE + tid;
    if (i0 >= n4) i0 = n4 - 1u;   // clamp (harmless garbage, store is guarded)
    __builtin_amdgcn_global_load_async_to_lds_b128(
        (__attribute__((address_space(1))) v4i*)(x + i0),
        (__attribute__((address_space(3))) v4i*)&buf[0][tid], 0, 0);
  }

  for (; t < ntiles; t += stride) {
    const unsigned tn = t + stride;
    const int nxt = cur ^ 1;
    if (tn < ntiles) {
      unsigned i1 = tn * TILE + tid;
      if (i1 >= n4) i1 = n4 - 1u;
      __builtin_amdgcn_global_load_async_to_lds_b128(
          (__attribute__((address_space(1))) v4i*)(x + i1),
          (__attribute__((address_space(3))) v4i*)&buf[nxt][tid], 0, 0);
      wait_async_le1();   // oldest (current tile) DMA complete
    } else {
      wait_async_le0();
    }

    v4f v = buf[cur][tid];
    const unsigned i = t * TILE + tid;
    if (i < n4) {
      v = rotate_pairs(v, i, cs, sn);
      __builtin_nontemporal_store(v, out + i);   // NT: don't pollute L2
    }
    cur = nxt;
  }
#else
  // --- Fallback: plain nontemporal b128 streaming -------------------------
  (void)ntiles;
  const unsigned gstride = gridDim.x * THREADS;
  for (unsigned i = blockIdx.x * THREADS + tid; i < n4; i += gstride) {
    v4f v = __builtin_nontemporal_load(x + i);
    v = rotate_pairs(v, i, cs, sn);
    __builtin_nontemporal_store(v, out + i);
  }
#endif
}

extern "C" void kernel_launch(void* const* d_in, const int* in_sizes, int n_in,
                              void* d_out, int out_size, void* d_ws, size_t ws_size,
                              hipStream_t stream) {
  (void)in_sizes; (void)n_in; (void)d_ws; (void)ws_size;
  const v4f*  x   = (const v4f*)d_in[0];    // (256,2048,25,2) f32, contiguous
  const float* w  = (const float*)d_in[1];  // (2048,1) f32; only w[0] is used
  v4f* out        = (v4f*)d_out;

  const unsigned n4     = (unsigned)(out_size / 4);        // 6,553,600 float4
  const unsigned ntiles = (n4 + TILE - 1u) / TILE;         // 25,600 tiles
  unsigned grid = ntiles < 3200u ? ntiles : 3200u;         // 8 tiles/block

  hipLaunchKernelGGL(rigid_rot_stream, dim3(grid), dim3(THREADS), 0, stream,
                     x, w, out, n4, ntiles);
}